// KANNetwork_30588757082466
// MI455X (gfx1250) — compile-verified
//
#include <hip/hip_runtime.h>

typedef __attribute__((ext_vector_type(2))) float v2f;
typedef __attribute__((ext_vector_type(8))) float v8f;

#define NINT 19   // NUM_KNOTS - 1

__device__ __forceinline__ float siluf(float v) {
  return v / (1.0f + __expf(-v));
}

// Per-wave 16-sample tile accumulator.
// WMMA f32 16x16x4 A layout: lanes 0-15 carry K=0,1 (=1, dx); lanes 16-31 carry
// K=2,3 (=dx^2, dx^3) for the same 16 samples. D = A x B accumulated over
// (channel i, interval k), A rows zeroed when sample's bucket != k. Exact f32.
template <int IN_D>
__device__ __forceinline__ v8f kan_accum(const float* __restrict__ sC,
                                         const float* __restrict__ hRow,
                                         bool lo, int n) {
  const float STEP  = 10.0f / 19.0f;
  const float ISTEP = 19.0f / 10.0f;

  float    a0[IN_D], a1[IN_D];
  int      idxv[IN_D];
  unsigned pm[IN_D];

  // Prologue: bucketize + power basis + per-lane bucket bit, all channels.
  #pragma unroll
  for (int i = 0; i < IN_D; ++i) {
    float hv  = hRow[i];
    int   idx = (int)floorf((hv + 5.0f) * ISTEP);   // searchsorted(right)-1
    idx = idx < 0 ? 0 : (idx > NINT - 1 ? NINT - 1 : idx);
    float dx  = hv - fmaf((float)idx, STEP, -5.0f);
    float dx2 = dx * dx, dx3 = dx2 * dx;
    a0[i]   = lo ? 1.0f : dx2;   // K = 0 or 2
    a1[i]   = lo ? dx   : dx3;   // K = 1 or 3
    idxv[i] = idx;
    pm[i]   = 1u << idx;
  }

  // Wave-wide OR-reduce of bucket masks: IN_D independent chains pipeline, so
  // only 4 dscnt waits total instead of 4 per channel.
  #pragma unroll
  for (int s = 1; s <= 8; s <<= 1) {
    #pragma unroll
    for (int i = 0; i < IN_D; ++i) pm[i] |= __shfl_xor(pm[i], s, 32);
  }

  v8f acc = {};
  #pragma unroll
  for (int i = 0; i < IN_D; ++i) {
    // scalar mask -> uniform branches, EXEC stays all-ones around WMMA
    unsigned rem = (unsigned)__builtin_amdgcn_readfirstlane((int)pm[i]);
    const float* cRow = sC + ((n * IN_D + i) * NINT) * 4 + (lo ? 0 : 2);
    if (!rem) continue;   // never taken (own bucket bit is set), keeps IR sane

    // Software-pipeline the B loads by one interval: issue load(k+1) before
    // the WMMA consuming load(k), so the wait relaxes to dscnt<=1 and the
    // LDS latency overlaps the previous matrix op.
    int kc = (int)__builtin_ctz(rem);
    rem &= rem - 1;
    v2f Bcur = *(const v2f*)(cRow + kc * 4);   // 8B-aligned ds_load_b64

    while (rem) {
      int kn = (int)__builtin_ctz(rem);
      rem &= rem - 1;
      v2f Bnext = *(const v2f*)(cRow + kn * 4);

      bool hit = (idxv[i] == kc);
      v2f A;
      A.x = hit ? a0[i] : 0.0f;
      A.y = hit ? a1[i] : 0.0f;
      acc = __builtin_amdgcn_wmma_f32_16x16x4_f32(
              false, A, false, Bcur, (short)0, acc, false, false);

      Bcur = Bnext;
      kc   = kn;
    }

    // epilogue: last interval
    {
      bool hit = (idxv[i] == kc);
      v2f A;
      A.x = hit ? a0[i] : 0.0f;
      A.y = hit ? a1[i] : 0.0f;
      acc = __builtin_amdgcn_wmma_f32_16x16x4_f32(
              false, A, false, Bcur, (short)0, acc, false, false);
    }
  }
  return acc;
}

__global__ __launch_bounds__(256)
void kan7_wmma_f32(const float* __restrict__ x, const float* __restrict__ t,
                   const float* __restrict__ c0, const float* __restrict__ b0,
                   const float* __restrict__ c1, const float* __restrict__ b1,
                   const float* __restrict__ c2, const float* __restrict__ b2,
                   const float* __restrict__ c3, const float* __restrict__ b3,
                   const float* __restrict__ c4, const float* __restrict__ b4,
                   const float* __restrict__ c5, const float* __restrict__ b5,
                   const float* __restrict__ c6, const float* __restrict__ b6,
                   float* __restrict__ out, int Btot)
{
  __shared__ __align__(16) float sC[16 * 16 * NINT * 4];  // 77824 B worst layer
  __shared__ float sBias[16];
  __shared__ __align__(16) float sH[8][16][20];           // 80B rows: aligned + conflict-free

  const float* cs[7] = {c0, c1, c2, c3, c4, c5, c6};
  const float* bs[7] = {b0, b1, b2, b3, b4, b5, b6};
  const int    ind[7]  = {3, 16, 16, 16, 16, 16, 16};
  const int    outd[7] = {16, 16, 16, 16, 16, 16, 2};

  const int  tid   = threadIdx.x;
  const int  lane  = tid & 31;
  const int  w     = tid >> 5;
  const bool lo    = lane < 16;
  const int  n     = lane & 15;
  const int  gbase = blockIdx.x * 128 + w * 16;

  const unsigned sCbase = (unsigned)(uintptr_t)(void*)sC;  // LDS byte address

  // ---- load input tile: h = [x0, x1, t] ----
  {
    int  s  = gbase + n;
    bool ok = s < Btot;
    if (lo) {
      sH[w][n][0] = ok ? x[s * 2 + 0] : 0.0f;
      sH[w][n][1] = ok ? x[s * 2 + 1] : 0.0f;
    } else {
      sH[w][n][2] = ok ? t[s] : 0.0f;
    }
  }

  for (int l = 0; l < 7; ++l) {
    const int in_d  = ind[l];
    const int out_d = outd[l];

    __syncthreads();  // everyone done reading sC from previous layer

    // ---- stage coefficients straight into LDS (no VGPR round-trip) ----
    // GLOBAL_LOAD_ASYNC_TO_LDS_B128, GV mode: full 64-bit address in a VGPR
    // pair, LDS destination byte address in a VGPR, tracked by ASYNCcnt.
    {
      const int nch = out_d * in_d * NINT;  // 16-byte chunks, contiguous
      unsigned long long src = (unsigned long long)(uintptr_t)cs[l];
      for (int cidx = tid; cidx < nch; cidx += 256) {
        unsigned           off      = (unsigned)cidx * 16u;
        unsigned           lds_addr = sCbase + off;
        unsigned long long gaddr    = src + (unsigned long long)off;
        asm volatile("global_load_async_to_lds_b128 %0, %1, off"
                     :: "v"(lds_addr), "v"(gaddr) : "memory");
      }
      if (tid < 16) sBias[tid] = (tid < out_d) ? bs[l][tid] : 0.0f;
      if (l < 6) __builtin_prefetch(cs[l + 1] + tid * 64, 0, 0);  // warm next layer
      asm volatile("s_wait_asynccnt 0x0" ::: "memory");
    }
    __syncthreads();

    v8f acc = (l == 0) ? kan_accum<3>(sC, &sH[w][n][0], lo, n)
                       : kan_accum<16>(sC, &sH[w][n][0], lo, n);

    // D layout: VGPR r, lanes 0-15 -> (M=r, N=lane); lanes 16-31 -> (M=r+8)
    float bias = sBias[n];
    if (l < 6) {
      #pragma unroll
      for (int r = 0; r < 8; ++r) {
        int m = r + (lo ? 0 : 8);
        sH[w][m][n] = siluf(acc[r] + bias);   // h for next layer
      }
    } else {
      if (n < 2) {                            // complex64: n=0 -> re, n=1 -> im
        #pragma unroll
        for (int r = 0; r < 8; ++r) {
          int s = gbase + r + (lo ? 0 : 8);
          if (s < Btot) out[s * 2 + n] = acc[r] + bias;
        }
      }
    }
  }
}

extern "C" void kernel_launch(void* const* d_in, const int* in_sizes, int n_in,
                              void* d_out, int out_size, void* d_ws, size_t ws_size,
                              hipStream_t stream) {
  const float* x = (const float*)d_in[0];
  const float* t = (const float*)d_in[1];
  const float* c[7];
  const float* b[7];
  for (int l = 0; l < 7; ++l) {
    c[l] = (const float*)d_in[2 + 2 * l];
    b[l] = (const float*)d_in[3 + 2 * l];
  }
  float* out = (float*)d_out;

  const int Btot   = in_sizes[0] / 2;       // x is [B, 2]
  const int blocks = (Btot + 127) / 128;    // 8 waves x 16 samples per block

  hipLaunchKernelGGL(kan7_wmma_f32, dim3(blocks), dim3(256), 0, stream,
                     x, t,
                     c[0], b[0], c[1], b[1], c[2], b[2], c[3], b[3],
                     c[4], b[4], c[5], b[5], c[6], b[6],
                     out, Btot);
}